// STU_25640954757239
// MI455X (gfx1250) — compile-verified
//
#include <hip/hip_runtime.h>
#include <math.h>

typedef __attribute__((ext_vector_type(2))) float v2f;
typedef __attribute__((ext_vector_type(8))) float v8f;

#define SEQ   2048
#define NFFT  4096
#define NF    2049          // rfft bins
#define FPAD  2112          // 33*64, padded bin count (rows >= NF are zero)
#define KF    24
#define DIM   768
#define JCH   (DIM / 16)    // 48 d-chunks per (sign,k)

// ---- CDNA5 async global->LDS helpers (ASYNCcnt-tracked) -------------------
__device__ __forceinline__ unsigned lds_addr_of(const void* p) {
    // flat pointer to __shared__ : low 32 bits are the LDS byte address
    return (unsigned)(unsigned long long)p;
}
__device__ __forceinline__ void async_ld_b128(unsigned lds, const float* base, unsigned goff) {
    asm volatile("global_load_async_to_lds_b128 %0, %1, %2"
                 :: "v"(lds), "v"(goff), "s"(base) : "memory");
}
__device__ __forceinline__ void wait_async0() {
    asm volatile("s_wait_asynccnt 0x0" ::: "memory");
}

// ---------------------------------------------------------------------------
// Kernel 0: twiddle table tw[i] = (cos(2*pi*i/4096), sin(2*pi*i/4096))
// ---------------------------------------------------------------------------
__global__ __launch_bounds__(256) void stu_gen_tw(float* __restrict__ tw) {
    int i = blockIdx.x * blockDim.x + threadIdx.x;
    if (i < NFFT) {
        float s, c;
        sincosf((float)(2.0 * M_PI * (double)i / (double)NFFT), &s, &c);
        tw[2 * i]     = c;
        tw[2 * i + 1] = s;
    }
}

// ---------------------------------------------------------------------------
// Kernel 1: rfft of x as direct DFT, exact integer phase idx = (f*s) mod 4096
// ---------------------------------------------------------------------------
__global__ __launch_bounds__(256) void stu_xdft(const float* __restrict__ x,
                                                const float* __restrict__ tw,
                                                float* __restrict__ Xr,
                                                float* __restrict__ Xi) {
    __shared__ float2 ltw[NFFT];
    int tid = threadIdx.x;
    for (int i = tid; i < NFFT; i += 256) ltw[i] = ((const float2*)tw)[i];
    __syncthreads();
    int f = blockIdx.x;
    int d = blockIdx.y * 256 + tid;
    float sr = 0.f, si = 0.f;
    if (f < NF) {
        int idx = 0;
        #pragma unroll 4
        for (int s = 0; s < SEQ; ++s) {
            float2 cs = ltw[idx];
            float  xv = x[s * DIM + d];
            sr  = fmaf(xv, cs.x, sr);
            si  = fmaf(xv, cs.y, si);
            idx = (idx + f) & (NFFT - 1);
        }
    }
    Xr[f * DIM + d] = sr;
    Xi[f * DIM + d] = -si;
}

// ---------------------------------------------------------------------------
// Kernel 2: spectra of phi (plus) and (-1)^t * phi (minus)
// ---------------------------------------------------------------------------
__global__ __launch_bounds__(32) void stu_phidft(const float* __restrict__ phi,
                                                 const float* __restrict__ tw,
                                                 float* __restrict__ PRp, float* __restrict__ PIp,
                                                 float* __restrict__ PRm, float* __restrict__ PIm) {
    int f = blockIdx.x;
    int k = threadIdx.x;
    if (k >= KF) return;
    float prp = 0.f, pip = 0.f, prm = 0.f, pim = 0.f;
    if (f < NF) {
        int idx = 0;
        for (int t = 0; t < SEQ; ++t) {
            float c = tw[2 * idx], s = tw[2 * idx + 1];
            float p  = phi[t * KF + k];
            float pm = (t & 1) ? -p : p;
            prp = fmaf(p,  c, prp);  pip = fmaf(p,  s, pip);
            prm = fmaf(pm, c, prm);  pim = fmaf(pm, s, pim);
            idx = (idx + f) & (NFFT - 1);
        }
    }
    PRp[f * KF + k] = prp;  PIp[f * KF + k] = -pip;
    PRm[f * KF + k] = prm;  PIm[f * KF + k] = -pim;
}

// ---------------------------------------------------------------------------
// Kernel 3 (WMMA + async double-buffered staging):
//   G[f,e] += Phi_sign[f,k] * (Xf[f,:] @ M_sign[k,:,:])   (complex, fused)
// Block = 128 threads (4 waves), tile 64(f) x 64(e). Nested (sgn,k,j) loops;
// each stage waits its async tiles, issues the next stage's transfers into
// the other LDS buffer, then runs 32 f32 WMMAs on the current buffer.
// W accumulators are zeroed once per k (outside the j loop).
// ---------------------------------------------------------------------------
__global__ __launch_bounds__(128) void stu_specmm(
        const float* __restrict__ Xr, const float* __restrict__ Xi,
        const float* __restrict__ Mp, const float* __restrict__ Mm,
        const float* __restrict__ PRp, const float* __restrict__ PIp,
        const float* __restrict__ PRm, const float* __restrict__ PIm,
        float* __restrict__ Gr, float* __restrict__ Gi) {
    __shared__ float lXr[2][64 * 16];
    __shared__ float lXi[2][64 * 16];
    __shared__ float lM [2][16 * 64];

    const int tid  = threadIdx.x;
    const int wave = tid >> 5;
    const int lane = tid & 31;
    const int hi   = lane >> 4;
    const int lo   = lane & 15;
    const int f0   = blockIdx.x * 64;
    const int e0   = blockIdx.y * 64;

    // Per-thread async transfer descriptors: 2 x B128 per tile per thread.
    unsigned xLds[2], xGoff[2], mLds[2], mGoff[2];
    #pragma unroll
    for (int a = 0; a < 2; ++a) {
        int c  = tid + a * 128;
        int xr = c >> 2, xc = (c & 3) * 4;           // X tile: 64 rows x 16 cols
        xLds[a]  = (unsigned)((xr * 16 + xc) * 4);
        xGoff[a] = (unsigned)(((f0 + xr) * DIM + xc) * 4);
        int mr = c >> 4, mc = (c & 15) * 4;          // M tile: 16 rows x 64 cols
        mLds[a]  = (unsigned)((mr * 64 + mc) * 4);
        mGoff[a] = (unsigned)((mr * DIM + e0 + mc) * 4);
    }
    const unsigned bXr = lds_addr_of(&lXr[0][0]);
    const unsigned bXi = lds_addr_of(&lXi[0][0]);
    const unsigned bM  = lds_addr_of(&lM[0][0]);

    const v8f vzero = {0.f, 0.f, 0.f, 0.f, 0.f, 0.f, 0.f, 0.f};
    v8f Gre[4], Gim[4];
    #pragma unroll
    for (int t = 0; t < 4; ++t) { Gre[t] = vzero; Gim[t] = vzero; }

    unsigned buf = 0;

    // Prologue: issue stage (sgn=0, k=0, j=0) into buffer 0.
    #pragma unroll
    for (int a = 0; a < 2; ++a) {
        async_ld_b128(bXr + xLds[a], Xr, xGoff[a]);
        async_ld_b128(bXi + xLds[a], Xi, xGoff[a]);
        async_ld_b128(bM  + mLds[a], Mp, mGoff[a]);
    }

    for (int sgn = 0; sgn < 2; ++sgn) {
        const float* PR = sgn ? PRm : PRp;
        const float* PI = sgn ? PIm : PIp;
        for (int k = 0; k < KF; ++k) {
            v8f Wr[4], Wi[4];
            #pragma unroll
            for (int t = 0; t < 4; ++t) { Wr[t] = vzero; Wi[t] = vzero; }

            for (int j = 0; j < JCH; ++j) {
                wait_async0();      // this wave's tiles for current stage landed
                __syncthreads();    // all waves landed + other buffer free

                // next-stage coordinates (simple carry logic, all uniform SALU)
                int nj = j + 1, nk = k, ns = sgn;
                if (nj == JCH) { nj = 0; ++nk; if (nk == KF) { nk = 0; ++ns; } }
                if (ns < 2) {
                    const float* Mb   = ns ? Mm : Mp;
                    const unsigned ob = (buf ^ 1u) * (unsigned)(64 * 16 * 4);
                    const unsigned xa = (unsigned)(nj * 16 * 4);
                    const unsigned ma = (unsigned)((nk * DIM * DIM + nj * 16 * DIM) * 4);
                    #pragma unroll
                    for (int a = 0; a < 2; ++a) {
                        async_ld_b128(bXr + ob + xLds[a], Xr, xGoff[a] + xa);
                        async_ld_b128(bXi + ob + xLds[a], Xi, xGoff[a] + xa);
                        async_ld_b128(bM  + ob + mLds[a], Mb, mGoff[a] + ma);
                    }
                }

                // 32 WMMAs on current buffer
                #pragma unroll
                for (int dd = 0; dd < 16; dd += 4) {
                    const int krow = dd + hi * 2;              // even => aligned v2f
                    v2f ar = *(const v2f*)&lXr[buf][(wave * 16 + lo) * 16 + krow];
                    v2f ai = *(const v2f*)&lXi[buf][(wave * 16 + lo) * 16 + krow];
                    #pragma unroll
                    for (int et = 0; et < 4; ++et) {
                        const int mb = krow * 64 + et * 16 + lo;
                        v2f b;
                        b.x = lM[buf][mb];
                        b.y = lM[buf][mb + 64];
                        Wr[et] = __builtin_amdgcn_wmma_f32_16x16x4_f32(
                                     false, ar, false, b, (short)0, Wr[et], false, false);
                        Wi[et] = __builtin_amdgcn_wmma_f32_16x16x4_f32(
                                     false, ai, false, b, (short)0, Wi[et], false, false);
                    }
                }
                buf ^= 1u;
            }

            // fused epilogue: G += Phi[f,k] * W  (complex), once per k
            #pragma unroll
            for (int jj = 0; jj < 8; ++jj) {
                const int f = f0 + wave * 16 + jj + hi * 8;
                const float pr = PR[f * KF + k];
                const float pi = PI[f * KF + k];
                #pragma unroll
                for (int et = 0; et < 4; ++et) {
                    Gre[et][jj] += pr * Wr[et][jj] - pi * Wi[et][jj];
                    Gim[et][jj] += pr * Wi[et][jj] + pi * Wr[et][jj];
                }
            }
        }
    }

    #pragma unroll
    for (int et = 0; et < 4; ++et)
        #pragma unroll
        for (int jj = 0; jj < 8; ++jj) {
            const int f = f0 + wave * 16 + jj + hi * 8;
            const int e = e0 + et * 16 + lo;
            Gr[f * DIM + e] = Gre[et][jj];
            Gi[f * DIM + e] = Gim[et][jj];
        }
}

// ---------------------------------------------------------------------------
// Kernel 4: irfft of G along f (half-spectrum, hermitian weights folded in)
// ---------------------------------------------------------------------------
__global__ __launch_bounds__(256) void stu_idft(const float* __restrict__ Gr,
                                                const float* __restrict__ Gi,
                                                const float* __restrict__ tw,
                                                float* __restrict__ out) {
    __shared__ float2 ltw[NFFT];
    int tid = threadIdx.x;
    for (int i = tid; i < NFFT; i += 256) ltw[i] = ((const float2*)tw)[i];
    __syncthreads();
    int s = blockIdx.x;
    int e = blockIdx.y * 256 + tid;
    float acc = 0.f;
    int idx = s & (NFFT - 1);
    #pragma unroll 4
    for (int f = 1; f < SEQ; ++f) {
        float2 cs = ltw[idx];
        acc = fmaf(Gr[f * DIM + e],  cs.x, acc);
        acc = fmaf(-Gi[f * DIM + e], cs.y, acc);
        idx = (idx + s) & (NFFT - 1);
    }
    float edge = Gr[e];                                     // f = 0
    edge += Gr[2048 * DIM + e] * ((s & 1) ? -1.f : 1.f);    // f = N/2 (sin = 0)
    out[s * DIM + e] = (edge + 2.f * acc) * (1.0f / NFFT);
}

// ---------------------------------------------------------------------------
extern "C" void kernel_launch(void* const* d_in, const int* in_sizes, int n_in,
                              void* d_out, int out_size, void* d_ws, size_t ws_size,
                              hipStream_t stream) {
    const float* x   = (const float*)d_in[0];   // (1, 2048, 768)
    const float* phi = (const float*)d_in[1];   // (2048, 24)
    const float* Mp  = (const float*)d_in[2];   // (24, 768, 768)
    const float* Mm  = (const float*)d_in[3];   // (24, 768, 768)
    float* out = (float*)d_out;                 // (1, 2048, 768)

    float* ws  = (float*)d_ws;                  // ~27 MB f32 scratch
    float* tw  = ws;                            // 2*4096
    float* Xr  = tw  + 2 * NFFT;                // FPAD*DIM
    float* Xi  = Xr  + FPAD * DIM;
    float* PRp = Xi  + FPAD * DIM;              // FPAD*KF each
    float* PIp = PRp + FPAD * KF;
    float* PRm = PIp + FPAD * KF;
    float* PIm = PRm + FPAD * KF;
    float* Gr  = PIm + FPAD * KF;               // FPAD*DIM each
    float* Gi  = Gr  + FPAD * DIM;

    stu_gen_tw<<<dim3(16), dim3(256), 0, stream>>>(tw);
    stu_xdft  <<<dim3(FPAD, 3), dim3(256), 0, stream>>>(x, tw, Xr, Xi);
    stu_phidft<<<dim3(FPAD), dim3(32), 0, stream>>>(phi, tw, PRp, PIp, PRm, PIm);
    stu_specmm<<<dim3(FPAD / 64, DIM / 64), dim3(128), 0, stream>>>(
        Xr, Xi, Mp, Mm, PRp, PIp, PRm, PIm, Gr, Gi);
    stu_idft  <<<dim3(SEQ, 3), dim3(256), 0, stream>>>(Gr, Gi, tw, out);
}